// Net_BiAttn2D_84954453115087
// MI455X (gfx1250) — compile-verified
//
#include <hip/hip_runtime.h>
#include <hip/hip_bf16.h>

// ---------------- problem constants (match reference) ----------------
constexpr int   NODES  = 100000;
constexpr int   EDGES  = 1600000;
constexpr int   GB     = 500;     // graphs (B)
constexpr int   D_IN   = 50;
constexpr int   H1D    = 100;
constexpr int   DGD    = 20;
constexpr int   PP     = 200;
constexpr int   DAD    = 64;
constexpr int   FIN    = (DGD + 1) * (DAD + 1);   // 1365
constexpr int   FINP   = (FIN + 3) & ~3;          // 1368 (pad K to mult of 4, even)
constexpr int   NPG    = NODES / GB;              // 200 nodes per graph
constexpr float LN_EPS   = 1e-5f;
constexpr float BN_EPS   = 1e-5f;
constexpr float DEG_EPS  = 1e-12f;
constexpr float ATTN_EPS = 1e-12f;

typedef __attribute__((ext_vector_type(2))) float v2f;
typedef __attribute__((ext_vector_type(8))) float v8f;

#define WMMA_F32(A_, B_, C_) \
  __builtin_amdgcn_wmma_f32_16x16x4_f32(false, (A_), false, (B_), (short)0, (C_), false, false)

// ---------------- B transpose (+K padding): Bt[n*Kp + k] = B[k*N + n] ----------
__global__ void transposeB(const float* __restrict__ B, float* __restrict__ Bt,
                           int K, int Kp, int N) {
  int i = blockIdx.x * blockDim.x + threadIdx.x;
  if (i >= N * Kp) return;
  int n = i / Kp, k = i % Kp;
  Bt[i] = (k < K) ? B[(size_t)k * N + n] : 0.f;
}

// one K-group of fragments (k..k+3 for this wave)
struct Frag { v2f a, b0, b1, b2, b3; };

// ---------------- fp32 WMMA GEMM: C(MxN) = A(MxK) @ B, B given as Bt(NxK) -----
// One wave computes a 16x64 strip (4 n-tiles) with V_WMMA_F32_16X16X4_F32.
// Software pipeline with prefetch distance 2: two fragment groups resident,
// group i+2 loads issue while group i feeds 4 independent WMMAs, keeping
// 10 b64 loads in flight so waits stay partial.
template <bool BIAS, bool RELU>
__global__ __launch_bounds__(128)
void gemm_wmma_f32(const float* __restrict__ A, const float* __restrict__ Bt,
                   const float* __restrict__ bias, float* __restrict__ C,
                   int M, int K, int N) {
  const int mt = (M + 15) >> 4;
  const int ng = (N + 63) >> 6;              // groups of 4 n-tiles
  const int wid  = threadIdx.x >> 5;
  const int tile = blockIdx.x * 4 + wid;
  if (tile >= mt * ng) return;
  const int tm = tile / ng;
  const int g  = tile % ng;
  const int lane = threadIdx.x & 31;
  const int half = lane >> 4;                // 0: K=(k0,k0+1), 1: K=(k0+2,k0+3)
  const int l    = lane & 15;

  int arow = tm * 16 + l;  if (arow > M - 1) arow = M - 1;
  const int col0 = g * 64 + l;               // actual store columns: col0 + j*16
  int cc0 = col0;       if (cc0 > N - 1) cc0 = N - 1;
  int cc1 = col0 + 16;  if (cc1 > N - 1) cc1 = N - 1;
  int cc2 = col0 + 32;  if (cc2 > N - 1) cc2 = N - 1;
  int cc3 = col0 + 48;  if (cc3 > N - 1) cc3 = N - 1;

  const float* __restrict__ pa  = A  + (size_t)arow * K + half * 2;
  const float* __restrict__ pb0 = Bt + (size_t)cc0  * K + half * 2;
  const float* __restrict__ pb1 = Bt + (size_t)cc1  * K + half * 2;
  const float* __restrict__ pb2 = Bt + (size_t)cc2  * K + half * 2;
  const float* __restrict__ pb3 = Bt + (size_t)cc3  * K + half * 2;

  auto load_group = [&](int k) -> Frag {
    Frag f;
    f.a  = *(const v2f*)(pa  + k);
    f.b0 = *(const v2f*)(pb0 + k);
    f.b1 = *(const v2f*)(pb1 + k);
    f.b2 = *(const v2f*)(pb2 + k);
    f.b3 = *(const v2f*)(pb3 + k);
    return f;
  };

  v8f acc0 = {}, acc1 = {}, acc2 = {}, acc3 = {};
  auto mul_group = [&](const Frag& f) {
    acc0 = WMMA_F32(f.a, f.b0, acc0);
    acc1 = WMMA_F32(f.a, f.b1, acc1);
    acc2 = WMMA_F32(f.a, f.b2, acc2);
    acc3 = WMMA_F32(f.a, f.b3, acc3);
  };

  const int K4 = K & ~3;
  const int groups = K4 >> 2;
  if (groups > 0) {
    Frag f0 = load_group(0);
    if (groups > 1) {
      Frag f1 = load_group(4);
      #pragma unroll 2
      for (int i = 0; i + 2 < groups; ++i) {
        Frag f2 = load_group((i + 2) * 4);   // distance-2 prefetch
        mul_group(f0);
        f0 = f1; f1 = f2;
      }
      mul_group(f0);
      mul_group(f1);
    } else {
      mul_group(f0);
    }
  }
  if (K4 < K) {                              // K % 4 in {1,2,3}, one guarded step
    const int ka = K4 + half * 2;
    Frag f;
    f.a = (v2f){0.f, 0.f}; f.b0 = f.a; f.b1 = f.a; f.b2 = f.a; f.b3 = f.a;
    if (ka < K) {
      f.a.x = pa[K4]; f.b0.x = pb0[K4]; f.b1.x = pb1[K4];
      f.b2.x = pb2[K4]; f.b3.x = pb3[K4];
    }
    if (ka + 1 < K) {
      f.a.y = pa[K4 + 1]; f.b0.y = pb0[K4 + 1]; f.b1.y = pb1[K4 + 1];
      f.b2.y = pb2[K4 + 1]; f.b3.y = pb3[K4 + 1];
    }
    mul_group(f);
  }

  // ---- store epilogue (guards only here) ----
  const int crow0 = tm * 16 + half * 8;
  v8f accs[4] = {acc0, acc1, acc2, acc3};
  #pragma unroll
  for (int j = 0; j < 4; ++j) {
    const int col = col0 + j * 16;
    if (col < N) {
      float bv = 0.f;
      if constexpr (BIAS) bv = bias[col];
      #pragma unroll
      for (int r = 0; r < 8; ++r) {
        const int cr = crow0 + r;
        if (cr < M) {
          float v = accs[j][r] + bv;
          if constexpr (RELU) v = v > 0.f ? v : 0.f;
          C[(size_t)cr * N + col] = v;
        }
      }
    }
  }
}

// ---------------- GCN helpers ----------------
__global__ void deg_kernel(const int* __restrict__ dst, float* __restrict__ deg, int e) {
  int i = blockIdx.x * blockDim.x + threadIdx.x;
  if (i < e) atomicAdd(&deg[dst[i]], 1.0f);
}

__global__ void dinv_kernel(const float* __restrict__ deg, float* __restrict__ dinv, int n) {
  int i = blockIdx.x * blockDim.x + threadIdx.x;
  if (i < n) {
    float d = deg[i];
    d = d > DEG_EPS ? d : DEG_EPS;
    dinv[i] = rsqrtf(d);
  }
}

// out[row,d] = in[row,d] * dinv[row]   (in-place safe)
__global__ void scale_rows(const float* __restrict__ in, const float* __restrict__ dinv,
                           float* __restrict__ out, long long total, int D) {
  long long i = (long long)blockIdx.x * blockDim.x + threadIdx.x;
  if (i < total) out[i] = in[i] * dinv[i / D];
}

// x2[dst[e], d] += x1[src[e], d]
__global__ void edge_scatter(const float* __restrict__ x1, const int* __restrict__ src,
                             const int* __restrict__ dst, float* __restrict__ x2,
                             long long total, int D) {
  long long i = (long long)blockIdx.x * blockDim.x + threadIdx.x;
  if (i < total) {
    int e = (int)(i / D);
    int d = (int)(i % D);
    atomicAdd(&x2[(size_t)dst[e] * D + d], x1[(size_t)src[e] * D + d]);
  }
}

// ---------------- pooling (contiguous 200-node graphs) ----------------
__global__ void pool_mean(const float* __restrict__ h, float* __restrict__ hg) {
  int i = blockIdx.x * blockDim.x + threadIdx.x;
  if (i >= GB * DGD) return;
  int b = i / DGD, d = i % DGD;
  const float* base = h + (size_t)b * NPG * DGD + d;
  float s = 0.f;
  for (int j = 0; j < NPG; ++j) s += base[(size_t)j * DGD];
  hg[i] = s * (1.0f / NPG);
}

// ---------------- layernorm over 64 (one wave per row, wave32) ----------------
__device__ __forceinline__ void ln64_store(float v0, float v1, int d0,
                                           const float* __restrict__ g,
                                           const float* __restrict__ bb,
                                           float* __restrict__ outrow) {
  float s  = v0 + v1;
  float sq = v0 * v0 + v1 * v1;
  #pragma unroll
  for (int m = 16; m >= 1; m >>= 1) {
    s  += __shfl_xor(s,  m, 32);
    sq += __shfl_xor(sq, m, 32);
  }
  float mean = s * (1.0f / 64.0f);
  float var  = sq * (1.0f / 64.0f) - mean * mean;
  float inv  = rsqrtf(var + LN_EPS);
  outrow[d0]     = (v0 - mean) * inv * g[d0]     + bb[d0];
  outrow[d0 + 1] = (v1 - mean) * inv * g[d0 + 1] + bb[d0 + 1];
}

__global__ void ln_rows64(const float* __restrict__ in, const float* __restrict__ g,
                          const float* __restrict__ bb, float* __restrict__ out, int rows) {
  int row = blockIdx.x * (blockDim.x >> 5) + (threadIdx.x >> 5);
  if (row >= rows) return;
  int lane = threadIdx.x & 31;
  int d0 = lane * 2;
  const float* r = in + (size_t)row * DAD;
  ln64_store(r[d0], r[d0 + 1], d0, g, bb, out + (size_t)row * DAD);
}

// k/v path: pre[b,p,d] = desc[b,p]*W[d] + bias[d] + Epos[p,d]; then LN64
__global__ void kv_pre_ln(const float* __restrict__ desc, const float* __restrict__ W,
                          const float* __restrict__ bias, const float* __restrict__ Epos,
                          const float* __restrict__ g, const float* __restrict__ bb,
                          float* __restrict__ out, int rows) {
  int row = blockIdx.x * (blockDim.x >> 5) + (threadIdx.x >> 5);
  if (row >= rows) return;
  int lane = threadIdx.x & 31;
  int d0 = lane * 2;
  int p = row % PP;
  float x = desc[row];
  float v0 = x * W[d0]     + bias[d0]     + Epos[(size_t)p * DAD + d0];
  float v1 = x * W[d0 + 1] + bias[d0 + 1] + Epos[(size_t)p * DAD + d0 + 1];
  ln64_store(v0, v1, d0, g, bb, out + (size_t)row * DAD);
}

// ---------------- attention (one block per graph) ----------------
__global__ void attn_kernel(const float* __restrict__ q2, const float* __restrict__ k2,
                            const float* __restrict__ v, float* __restrict__ c2) {
  __shared__ float sq[DAD];
  __shared__ float sa[PP];
  __shared__ float sinv;
  const int b = blockIdx.x;
  const int t = threadIdx.x;
  if (t < DAD) sq[t] = q2[(size_t)b * DAD + t];
  __syncthreads();
  for (int p = t; p < PP; p += blockDim.x) {
    const float* kr = k2 + ((size_t)b * PP + p) * DAD;
    float e = 0.f;
    #pragma unroll 8
    for (int d = 0; d < DAD; ++d) e += kr[d] * sq[d];
    e *= 0.125f;                               // 1/sqrt(64)
    sa[p] = 1.0f / (1.0f + expf(-e));          // sigmoid
  }
  __syncthreads();
  if (t == 0) {
    float s = 0.f;
    for (int p = 0; p < PP; ++p) s += sa[p];
    sinv = 1.0f / (s + ATTN_EPS);
  }
  __syncthreads();
  for (int d = t; d < DAD; d += blockDim.x) {
    float acc = 0.f;
    for (int p = 0; p < PP; ++p) acc += sa[p] * v[((size_t)b * PP + p) * DAD + d];
    c2[(size_t)b * DAD + d] = acc * sinv;
  }
}

// ---------------- bilinear fusion: outer([hg,1],[c2,1]), stride padded -------
__global__ void fusion_kernel(const float* __restrict__ hg, const float* __restrict__ c2,
                              float* __restrict__ f) {
  int i = blockIdx.x * blockDim.x + threadIdx.x;
  if (i >= GB * FINP) return;
  int b = i / FINP, r = i % FINP;
  if (r >= FIN) { f[i] = 0.f; return; }
  int a = r / (DAD + 1), c = r % (DAD + 1);
  float av = (a < DGD) ? hg[(size_t)b * DGD + a] : 1.0f;
  float cv = (c < DAD) ? c2[(size_t)b * DAD + c] : 1.0f;
  f[i] = av * cv;
}

// ---------------- inference batchnorm + relu (in place) ----------------
__global__ void bn_relu(float* __restrict__ x, const float* __restrict__ g,
                        const float* __restrict__ bb, const float* __restrict__ m,
                        const float* __restrict__ v, int total, int C) {
  int i = blockIdx.x * blockDim.x + threadIdx.x;
  if (i >= total) return;
  int c = i % C;
  float y = (x[i] - m[c]) * rsqrtf(v[c] + BN_EPS) * g[c] + bb[c];
  x[i] = y > 0.f ? y : 0.f;
}

// ---------------- final 32->1 head ----------------
__global__ void final_head(const float* __restrict__ o2, const float* __restrict__ W3,
                           const float* __restrict__ b3, float* __restrict__ out) {
  int b = blockIdx.x * blockDim.x + threadIdx.x;
  if (b >= GB) return;
  float s = b3[0];
  #pragma unroll
  for (int i = 0; i < 32; ++i) s += o2[(size_t)b * 32 + i] * W3[i];
  out[b] = s;
}

// ---------------- host orchestration ----------------
static inline int cdiv(long long a, int b) { return (int)((a + b - 1) / b); }

template <bool BIAS, bool RELU>
static inline void gemm(const float* A, const float* B, float* Bt, const float* bias,
                        float* C, int M, int K, int N, hipStream_t s, int Kfull = -1) {
  // transpose (and pad) B into Bt, then run the WMMA GEMM with K = padded K
  const int Korig = (Kfull < 0) ? K : Kfull;       // rows of original B
  transposeB<<<cdiv((long long)N * K, 256), 256, 0, s>>>(B, Bt, Korig, K, N);
  long long tiles = (long long)((M + 15) / 16) * ((N + 63) / 64);
  gemm_wmma_f32<BIAS, RELU><<<cdiv(tiles, 4), 128, 0, s>>>(A, Bt, bias, C, M, K, N);
}

extern "C" void kernel_launch(void* const* d_in, const int* in_sizes, int n_in,
                              void* d_out, int out_size, void* d_ws, size_t ws_size,
                              hipStream_t stream) {
  const float* feat = (const float*)d_in[0];
  const int*   src  = (const int*)d_in[1];
  const int*   dst  = (const int*)d_in[2];
  // d_in[3] node_graph: contiguous blocks of NPG, handled analytically
  const float* desc = (const float*)d_in[4];
  const float* Wg1  = (const float*)d_in[5];
  const float* Wg2  = (const float*)d_in[6];
  const float* Wq   = (const float*)d_in[7];
  const float* bq   = (const float*)d_in[8];
  const float* Wk   = (const float*)d_in[9];
  const float* bk   = (const float*)d_in[10];
  const float* Wv   = (const float*)d_in[11];
  const float* bv   = (const float*)d_in[12];
  const float* Ek   = (const float*)d_in[13];
  const float* Ev   = (const float*)d_in[14];
  const float* Uq   = (const float*)d_in[15];
  const float* Vkm  = (const float*)d_in[16];
  const float* lnqg = (const float*)d_in[17];
  const float* lnqb = (const float*)d_in[18];
  const float* lnkg = (const float*)d_in[19];
  const float* lnkb = (const float*)d_in[20];
  const float* lnvg = (const float*)d_in[21];
  const float* lnvb = (const float*)d_in[22];
  const float* W1   = (const float*)d_in[23];
  const float* b1   = (const float*)d_in[24];
  const float* W2   = (const float*)d_in[25];
  const float* b2   = (const float*)d_in[26];
  const float* W3   = (const float*)d_in[27];
  const float* b3   = (const float*)d_in[28];
  const float* bn1g = (const float*)d_in[29];
  const float* bn1b = (const float*)d_in[30];
  const float* bn1m = (const float*)d_in[31];
  const float* bn1v = (const float*)d_in[32];
  const float* bn2g = (const float*)d_in[33];
  const float* bn2b = (const float*)d_in[34];
  const float* bn2m = (const float*)d_in[35];
  const float* bn2v = (const float*)d_in[36];

  // ---- carve workspace ----
  float* p = (float*)d_ws;
  float* deg  = p; p += NODES;
  float* dinv = p; p += NODES;
  float* bufA = p; p += (size_t)NODES * H1D;   // x1 / h2
  float* bufB = p; p += (size_t)NODES * H1D;   // x2 (accumulator) / GEMM input
  float* h1   = p; p += (size_t)NODES * H1D;
  float* hg   = p; p += GB * DGD;
  float* qpre = p; p += GB * DAD;
  float* q    = p; p += GB * DAD;
  float* q2   = p; p += GB * DAD;
  float* kb   = p; p += (size_t)GB * PP * DAD;
  float* vb   = p; p += (size_t)GB * PP * DAD;
  float* k2   = p; p += (size_t)GB * PP * DAD;
  float* c2   = p; p += GB * DAD;
  float* fus  = p; p += (size_t)GB * FINP;
  float* o1   = p; p += GB * 128;
  float* o2   = p; p += GB * 32;
  float* Bt   = p; p += (size_t)128 * FINP;    // transposed-weight scratch (max)

  const int T = 256;

  // ---- degree + normalization ----
  hipMemsetAsync(deg, 0, (size_t)NODES * sizeof(float), stream);
  deg_kernel<<<cdiv(EDGES, T), T, 0, stream>>>(dst, deg, EDGES);
  dinv_kernel<<<cdiv(NODES, T), T, 0, stream>>>(deg, dinv, NODES);

  // ---- GCN layer 1: feat(N x 50) -> h1(N x 100) ----
  {
    long long nd = (long long)NODES * D_IN;
    long long ed = (long long)EDGES * D_IN;
    scale_rows<<<cdiv(nd, T), T, 0, stream>>>(feat, dinv, bufA, nd, D_IN);
    hipMemsetAsync(bufB, 0, (size_t)nd * sizeof(float), stream);
    edge_scatter<<<cdiv(ed, T), T, 0, stream>>>(bufA, src, dst, bufB, ed, D_IN);
    scale_rows<<<cdiv(nd, T), T, 0, stream>>>(bufB, dinv, bufB, nd, D_IN);
    gemm<false, true>(bufB, Wg1, Bt, nullptr, h1, NODES, D_IN, H1D, stream);
  }

  // ---- GCN layer 2: h1(N x 100) -> h2(N x 20) in bufA ----
  {
    long long nd = (long long)NODES * H1D;
    long long ed = (long long)EDGES * H1D;
    scale_rows<<<cdiv(nd, T), T, 0, stream>>>(h1, dinv, bufA, nd, H1D);
    hipMemsetAsync(bufB, 0, (size_t)nd * sizeof(float), stream);
    edge_scatter<<<cdiv(ed, T), T, 0, stream>>>(bufA, src, dst, bufB, ed, H1D);
    scale_rows<<<cdiv(nd, T), T, 0, stream>>>(bufB, dinv, bufB, nd, H1D);
    gemm<false, true>(bufB, Wg2, Bt, nullptr, bufA, NODES, H1D, DGD, stream);
  }

  // ---- mean pooling ----
  pool_mean<<<cdiv(GB * DGD, T), T, 0, stream>>>(bufA, hg);

  // ---- q / k / v with layernorm ----
  gemm<true, false>(hg, Wq, Bt, bq, qpre, GB, DGD, DAD, stream);
  ln_rows64<<<cdiv(GB, 8), 256, 0, stream>>>(qpre, lnqg, lnqb, q, GB);
  kv_pre_ln<<<cdiv(GB * PP, 8), 256, 0, stream>>>(desc, Wk, bk, Ek, lnkg, lnkb, kb, GB * PP);
  kv_pre_ln<<<cdiv(GB * PP, 8), 256, 0, stream>>>(desc, Wv, bv, Ev, lnvg, lnvb, vb, GB * PP);

  // ---- projections ----
  gemm<false, false>(q, Uq, Bt, nullptr, q2, GB, DAD, DAD, stream);
  gemm<false, false>(kb, Vkm, Bt, nullptr, k2, GB * PP, DAD, DAD, stream);

  // ---- sigmoid attention ----
  attn_kernel<<<GB, 256, 0, stream>>>(q2, k2, vb, c2);

  // ---- bilinear fusion + MLP head ----
  fusion_kernel<<<cdiv((long long)GB * FINP, T), T, 0, stream>>>(hg, c2, fus);
  gemm<true, false>(fus, W1, Bt, b1, o1, GB, FINP, 128, stream, FIN);
  bn_relu<<<cdiv(GB * 128, T), T, 0, stream>>>(o1, bn1g, bn1b, bn1m, bn1v, GB * 128, 128);
  gemm<true, false>(o1, W2, Bt, b2, o2, GB, 128, 32, stream);
  bn_relu<<<cdiv(GB * 32, T), T, 0, stream>>>(o2, bn2g, bn2b, bn2m, bn2v, GB * 32, 32);
  final_head<<<cdiv(GB, T), T, 0, stream>>>(o2, W3, b3, (float*)d_out);
}